// CannyEdgeDetector3D_88622355186022
// MI455X (gfx1250) — compile-verified
//
#include <hip/hip_runtime.h>
#include <hip/hip_bf16.h>

// ---------------- problem constants ----------------
#define VOL_N 256
#define TILE  8          // output tile edge
#define XT    12         // TILE + 2*2 halo (raw / mag tile)
#define ST    10         // TILE + 2*1 halo (smoothed / nms tile)

#define HI_T 0.2f
#define LO_T 0.1f
// separable normalized Gaussian taps: w = {e^-.5, 1, e^-.5} / (1 + 2e^-.5)
#define GW0 0.27406862f
#define GW1 0.45186276f

// ---------------- gfx1250 async global->LDS path ----------------
#if defined(__AMDGCN__) && __has_builtin(__builtin_amdgcn_global_load_async_to_lds_b32) && __has_builtin(__builtin_amdgcn_s_wait_asynccnt)
#define ASYNC_LDS 1
#else
#define ASYNC_LDS 0
#endif

__device__ __forceinline__ void lds_async_b32(float* lds_dst, const float* gsrc) {
#if ASYNC_LDS
  __builtin_amdgcn_global_load_async_to_lds_b32(
      (__attribute__((address_space(1))) int*)(gsrc),
      (__attribute__((address_space(3))) int*)(lds_dst),
      /*offset=*/0, /*cpol=*/0);
#else
  *lds_dst = *gsrc;
#endif
}

__device__ __forceinline__ void lds_async_wait0() {
#if ASYNC_LDS
  __builtin_amdgcn_s_wait_asynccnt(0);
#endif
}

// reflect-pad-1 semantics (only indices -1 and 256 reflect); clamp the
// never-consumed halo corners (-2, 257) into range.
__device__ __forceinline__ int refl_clamp(int i) {
  i = (i == -1) ? 1 : i;
  i = (i == VOL_N) ? (VOL_N - 2) : i;
  i = i < 0 ? 0 : i;
  i = i > (VOL_N - 1) ? (VOL_N - 1) : i;
  return i;
}

// =====================================================================
// Kernel A: Gaussian smooth (separable, via LDS) + Sobel + |grad|
// writes mag volume with zero border (matches jnp.pad(mag, 1)).
// =====================================================================
__global__ __launch_bounds__(256) void canny_mag_kernel(const float* __restrict__ x,
                                                        float* __restrict__ mag) {
  __shared__ float xs[XT * XT * XT];   // raw tile, halo 2        (6.9 KB)
  __shared__ float t1[XT * XT * ST];   // after W conv            (5.8 KB)
  __shared__ float t2[XT * ST * ST];   // after H conv            (4.8 KB)
  __shared__ float ss[ST * ST * ST];   // smoothed, halo 1        (4.0 KB)

  const int t  = threadIdx.x;
  const int bx = blockIdx.x * TILE;    // W origin
  const int by = blockIdx.y * TILE;    // H origin
  const int bz = blockIdx.z * TILE;    // D origin

  // ---- async-stage the 12^3 input tile into LDS ----
  // coords maintained incrementally: 256 = 1*144 + 9*12 + 4
  {
    int lw = t % XT, lh = (t / XT) % XT, ld = t / (XT * XT);
    for (int l = t; l < XT * XT * XT; l += 256) {
      const int gw = refl_clamp(bx + lw - 2);
      const int gh = refl_clamp(by + lh - 2);
      const int gd = refl_clamp(bz + ld - 2);
      const size_t gi = ((size_t)gd * VOL_N + gh) * VOL_N + gw;
      lds_async_b32(&xs[l], x + gi);
      lw += 4;  if (lw >= XT) { lw -= XT; ++lh; }
      lh += 9;  if (lh >= XT) { lh -= XT; ++ld; }
      ld += 1;
    }
  }
  lds_async_wait0();
  __syncthreads();

  // ---- separable Gaussian: W pass (d:12, h:12, w:10); 256 = 2*120+1*10+6
  {
    int lw = t % ST, lh = (t / ST) % XT, ld = t / (ST * XT);
    for (int l = t; l < XT * XT * ST; l += 256) {
      const float* p = &xs[(ld * XT + lh) * XT + lw];
      t1[l] = GW0 * p[0] + GW1 * p[1] + GW0 * p[2];
      lw += 6;  if (lw >= ST) { lw -= ST; ++lh; }
      lh += 1;  if (lh >= XT) { lh -= XT; ++ld; }
      ld += 2;
    }
  }
  __syncthreads();
  // ---- H pass (d:12, h:10, w:10); 256 = 2*100+5*10+6
  {
    int lw = t % ST, lh = (t / ST) % ST, ld = t / (ST * ST);
    for (int l = t; l < XT * ST * ST; l += 256) {
      const float* p = &t1[(ld * XT + lh) * ST + lw];
      t2[l] = GW0 * p[0] + GW1 * p[ST] + GW0 * p[2 * ST];
      lw += 6;  if (lw >= ST) { lw -= ST; ++lh; }
      lh += 5;  if (lh >= ST) { lh -= ST; ++ld; }
      ld += 2;
    }
  }
  __syncthreads();
  // ---- D pass (10^3); 256 = 2*100+5*10+6
  {
    int lw = t % ST, lh = (t / ST) % ST, ld = t / (ST * ST);
    for (int l = t; l < ST * ST * ST; l += 256) {
      const float* p = &t2[(ld * ST + lh) * ST + lw];
      ss[l] = GW0 * p[0] + GW1 * p[ST * ST] + GW0 * p[2 * ST * ST];
      lw += 6;  if (lw >= ST) { lw -= ST; ++lh; }
      lh += 5;  if (lh >= ST) { lh -= ST; ++ld; }
      ld += 2;
    }
  }
  __syncthreads();

  // ---- factored Sobel + magnitude (8^3; 256 = 4 planes exactly) ----
  {
    const int lw = t & 7;
    const int lh = (t >> 3) & 7;
    int ld = t >> 6;
    const int gw = bx + lw, gh = by + lh;
    #pragma unroll
    for (int it = 0; it < 2; ++it, ld += 4) {
      const int gd = bz + ld;
      float m = 0.0f;
      if (gw >= 1 && gw <= VOL_N - 2 && gh >= 1 && gh <= VOL_N - 2 &&
          gd >= 1 && gd <= VOL_N - 2) {
        float za[3][3], zd[3][3];   // sum / diff over depth axis
        #pragma unroll
        for (int b = 0; b < 3; ++b) {
          #pragma unroll
          for (int c = 0; c < 3; ++c) {
            const float v0 = ss[((ld + 0) * ST + (lh + b)) * ST + (lw + c)];
            const float v1 = ss[((ld + 1) * ST + (lh + b)) * ST + (lw + c)];
            const float v2 = ss[((ld + 2) * ST + (lh + b)) * ST + (lw + c)];
            za[b][c] = v0 + v1 + v2;
            zd[b][c] = v2 - v0;
          }
        }
        const float gx = (za[0][2] - za[0][0]) + 2.0f * (za[1][2] - za[1][0]) + (za[2][2] - za[2][0]);
        const float gy = (za[2][0] - za[0][0]) + 2.0f * (za[2][1] - za[0][1]) + (za[2][2] - za[0][2]);
        const float gz = zd[0][0] + zd[0][1] + zd[0][2] +
                         zd[1][0] + zd[1][1] + zd[1][2] +
                         zd[2][0] + zd[2][1] + zd[2][2];
        m = sqrtf(gx * gx + gy * gy + gz * gz);
      }
      mag[((size_t)gd * VOL_N + gh) * VOL_N + gw] = m;
    }
  }
}

// =====================================================================
// Kernel B: NMS (8 'upper' neighbors, strictly greater, interior only)
// + single-pass hysteresis.  mag re-reads are L2 hits (64MB < 192MB L2).
// =====================================================================
__global__ __launch_bounds__(256) void canny_nms_hyst_kernel(const float* __restrict__ mag,
                                                             float* __restrict__ out) {
  __shared__ float ms[XT * XT * XT];   // mag tile, halo 2
  __shared__ float ns[ST * ST * ST];   // nms tile, halo 1

  const int t  = threadIdx.x;
  const int bx = blockIdx.x * TILE;
  const int by = blockIdx.y * TILE;
  const int bz = blockIdx.z * TILE;

  // ---- stage mag tile (12^3); OOB -> 0 (feeds only zeroed border lanes)
  {
    int lw = t % XT, lh = (t / XT) % XT, ld = t / (XT * XT);
    for (int l = t; l < XT * XT * XT; l += 256) {
      const int gw = bx + lw - 2;
      const int gh = by + lh - 2;
      const int gd = bz + ld - 2;
      if ((unsigned)gw < (unsigned)VOL_N && (unsigned)gh < (unsigned)VOL_N &&
          (unsigned)gd < (unsigned)VOL_N) {
        const size_t gi = ((size_t)gd * VOL_N + gh) * VOL_N + gw;
        lds_async_b32(&ms[l], mag + gi);
      } else {
        ms[l] = 0.0f;
      }
      lw += 4;  if (lw >= XT) { lw -= XT; ++lh; }
      lh += 9;  if (lh >= XT) { lh -= XT; ++ld; }
      ld += 1;
    }
  }
  lds_async_wait0();
  __syncthreads();

  // ---- NMS into halo-1 LDS tile (10^3) ----
  {
    int lw = t % ST, lh = (t / ST) % ST, ld = t / (ST * ST);
    for (int l = t; l < ST * ST * ST; l += 256) {
      const int gw = bx + lw - 1;
      const int gh = by + lh - 1;
      const int gd = bz + ld - 1;
      const float cv = ms[((ld + 1) * XT + (lh + 1)) * XT + (lw + 1)];
      bool keep = (gw >= 1 && gw <= VOL_N - 2 && gh >= 1 && gh <= VOL_N - 2 &&
                   gd >= 1 && gd <= VOL_N - 2);
      // 8 neighbors at depth offset -1, excluding (-1,0,0)
      #pragma unroll
      for (int dh = 0; dh < 3; ++dh) {
        #pragma unroll
        for (int dw = 0; dw < 3; ++dw) {
          if (dh == 1 && dw == 1) continue;
          keep = keep && (cv > ms[(ld * XT + (lh + dh)) * XT + (lw + dw)]);
        }
      }
      ns[l] = keep ? cv : 0.0f;
      lw += 6;  if (lw >= ST) { lw -= ST; ++lh; }
      lh += 5;  if (lh >= ST) { lh -= ST; ++ld; }
      ld += 2;
    }
  }
  __syncthreads();

  // ---- thresholds + single-pass hysteresis (8^3; 4 planes/iter) ----
  {
    const int lw = t & 7;
    const int lh = (t >> 3) & 7;
    int ld = t >> 6;
    const int gw = bx + lw, gh = by + lh;
    #pragma unroll
    for (int it = 0; it < 2; ++it, ld += 4) {
      const int gd = bz + ld;
      const float nv = ns[((ld + 1) * ST + (lh + 1)) * ST + (lw + 1)];
      const bool strong   = nv > HI_T;
      const bool weak     = (nv > LO_T) && !strong;
      const bool interior = (gw >= 1 && gw <= VOL_N - 2 && gh >= 1 && gh <= VOL_N - 2 &&
                             gd >= 1 && gd <= VOL_N - 2);
      const bool any_strong =
          (ns[((ld + 0) * ST + (lh + 1)) * ST + (lw + 1)] > HI_T) ||
          (ns[((ld + 2) * ST + (lh + 1)) * ST + (lw + 1)] > HI_T) ||
          (ns[((ld + 1) * ST + (lh + 0)) * ST + (lw + 1)] > HI_T) ||
          (ns[((ld + 1) * ST + (lh + 2)) * ST + (lw + 1)] > HI_T) ||
          (ns[((ld + 1) * ST + (lh + 1)) * ST + (lw + 0)] > HI_T) ||
          (ns[((ld + 1) * ST + (lh + 1)) * ST + (lw + 2)] > HI_T);
      const bool conn = strong || (weak && any_strong && interior);
      // streamed result, never re-read: keep it out of L2 (x/mag stay resident)
      __builtin_nontemporal_store(conn ? 1.0f : 0.0f,
                                  &out[((size_t)gd * VOL_N + gh) * VOL_N + gw]);
    }
  }
}

// =====================================================================
extern "C" void kernel_launch(void* const* d_in, const int* in_sizes, int n_in,
                              void* d_out, int out_size, void* d_ws, size_t ws_size,
                              hipStream_t stream) {
  (void)in_sizes; (void)n_in; (void)out_size; (void)ws_size;
  const float* x   = (const float*)d_in[0];
  float*       out = (float*)d_out;
  float*       mag = (float*)d_ws;      // 256^3 floats = 64 MB scratch

  dim3 block(256);
  dim3 grid(VOL_N / TILE, VOL_N / TILE, VOL_N / TILE);   // 32^3 tiles
  canny_mag_kernel<<<grid, block, 0, stream>>>(x, mag);
  canny_nms_hyst_kernel<<<grid, block, 0, stream>>>(mag, out);
}